// TritonHybridBlock_13219909337136
// MI455X (gfx1250) — compile-verified
//
#include <hip/hip_runtime.h>
#include <stdint.h>

// Fused LNN/SNN hybrid scan:  h = a_t*h + b_t;  s = (h > thr);  h = s ? 0 : h
// Memory-bound streaming scan (0.25 FLOP/byte; roofline ~23us @ 23.3 TB/s).
// CDNA5 paths: global_load_async_to_lds_b128 (ASYNCcnt DMA), s_wait_asynccnt,
// non-temporal output stores. 512 single-wave workgroups, quad-buffered LDS
// tiles, prefetch depth 3.

#define CHUNK 32   // time steps per tile
#define NBUF  4    // LDS buffers (prefetch depth 3)

__global__ __launch_bounds__(32) void hybrid_scan_kernel(
    const float* __restrict__ a, const float* __restrict__ b,
    const float* __restrict__ thr, const float* __restrict__ h0,
    float* __restrict__ out_h, float* __restrict__ out_s,
    int T, int D)
{
    // [buf][t][32 lanes] tiles, 4 KB per matrix per buffer -> 32 KB total
    __shared__ __align__(16) float shA[NBUF * CHUNK * 32];
    __shared__ __align__(16) float shB[NBUF * CHUNK * 32];

    const int lane  = threadIdx.x & 31;
    const int dbase = (int)blockIdx.x << 5;   // 32 d-lanes per wave
    const int b_idx = (int)blockIdx.y;

    // Uniform (SGPR) global bases for this wave's (b, d-window)
    const float* gA = a + ((size_t)b_idx * (size_t)T) * (size_t)D + dbase;
    const float* gB = b + ((size_t)b_idx * (size_t)T) * (size_t)D + dbase;

    // One b128 async op copies 16B/lane = 512B = 4 time-rows of 128B:
    //   lane -> row (lane>>3), byte sub-offset (lane&7)*16
    const int      tl     = lane >> 3;
    const int      xo     = (lane & 7) << 4;
    const unsigned lane_g = (unsigned)(tl * D * 4 + xo);     // global byte off
    const unsigned lane_l = (unsigned)(tl * (32 * 4) + xo);  // LDS byte off

    // Low 32 bits of a generic pointer to __shared__ == LDS byte address
    const unsigned ldsA0 = (unsigned)(uintptr_t)(void*)shA;
    const unsigned ldsB0 = (unsigned)(uintptr_t)(void*)shB;

    float       h    = h0[(size_t)b_idx * (size_t)D + dbase + lane];
    const float th_v = thr[dbase + lane];

    const int nchunk = T / CHUNK;

    auto issue = [&](int c) {
        const int      buf = c & (NBUF - 1);
        const unsigned cg  = (unsigned)c * (unsigned)(CHUNK * D * 4);
        const unsigned la  = ldsA0 + (unsigned)(buf * CHUNK * 32 * 4) + lane_l;
        const unsigned lb  = ldsB0 + (unsigned)(buf * CHUNK * 32 * 4) + lane_l;
#pragma unroll
        for (int j = 0; j < (CHUNK / 4); ++j) {              // 8 ops / matrix
            const unsigned voff = cg + (unsigned)(j * 4 * D * 4) + lane_g;
            const unsigned lo   = (unsigned)(j * 512);
            asm volatile("global_load_async_to_lds_b128 %0, %1, %2"
                         :: "v"(la + lo), "v"(voff), "s"(gA) : "memory");
            asm volatile("global_load_async_to_lds_b128 %0, %1, %2"
                         :: "v"(lb + lo), "v"(voff), "s"(gB) : "memory");
        }
    };

    auto compute = [&](int c) {
        const int buf  = c & (NBUF - 1);
        const int base = buf * CHUNK * 32 + lane;
        size_t idx = ((size_t)b_idx * (size_t)T + (size_t)c * CHUNK) * (size_t)D
                     + (size_t)(dbase + lane);
#pragma unroll
        for (int t = 0; t < CHUNK; ++t) {
            const float av = shA[base + t * 32];   // ds_load_b32, conflict-free
            const float bv = shB[base + t * 32];
            const float hp = __builtin_fmaf(av, h, bv);
            const bool  sp = hp > th_v;            // m = hp - thr > 0
            __builtin_nontemporal_store(sp ? 1.0f : 0.0f, &out_s[idx]);
            h = sp ? 0.0f : hp;                    // hard reset
            __builtin_nontemporal_store(h, &out_h[idx]);
            idx += (size_t)D;
        }
    };

    if (nchunk > 0) issue(0);
    if (nchunk > 1) issue(1);
    if (nchunk > 2) issue(2);

    // Steady state: constant-depth pipeline, no branches around the wait.
    int c = 0;
    for (; c + 3 < nchunk; ++c) {
        issue(c + 3);
        asm volatile("s_wait_asynccnt 48" ::: "memory");  // retire oldest chunk
        compute(c);
    }
    // Drain: at most 3 chunks left in flight.
    for (; c < nchunk; ++c) {
        const int pend = nchunk - 1 - c;                  // 2, 1, 0
        if (pend == 2)      asm volatile("s_wait_asynccnt 32" ::: "memory");
        else if (pend == 1) asm volatile("s_wait_asynccnt 16" ::: "memory");
        else                asm volatile("s_wait_asynccnt 0"  ::: "memory");
        compute(c);
    }

    // Tail (T not a multiple of CHUNK): direct global loads
    for (int t = nchunk * CHUNK; t < T; ++t) {
        const float av = gA[(size_t)t * D + lane];
        const float bv = gB[(size_t)t * D + lane];
        const float hp = __builtin_fmaf(av, h, bv);
        const bool  sp = hp > th_v;
        const size_t idx = ((size_t)b_idx * T + t) * (size_t)D + dbase + lane;
        __builtin_nontemporal_store(sp ? 1.0f : 0.0f, &out_s[idx]);
        h = sp ? 0.0f : hp;
        __builtin_nontemporal_store(h, &out_h[idx]);
    }
}

extern "C" void kernel_launch(void* const* d_in, const int* in_sizes, int n_in,
                              void* d_out, int out_size, void* d_ws, size_t ws_size,
                              hipStream_t stream) {
    const float* a   = (const float*)d_in[0];   // (B,T,D)
    const float* b   = (const float*)d_in[1];   // (B,T,D)
    const float* thr = (const float*)d_in[2];   // (D,)
    const float* h0  = (const float*)d_in[3];   // (B,D)

    const int D  = in_sizes[2];
    const int BD = in_sizes[3];
    const int B  = BD / D;
    const int T  = in_sizes[0] / BD;

    float* out_h = (float*)d_out;                             // (B,T,D)
    float* out_s = out_h + (size_t)B * (size_t)T * (size_t)D; // (B,T,D)

    dim3 grid((unsigned)(D / 32), (unsigned)B);  // no int-div in kernel
    dim3 block(32);
    hipLaunchKernelGGL(hybrid_scan_kernel, grid, block, 0, stream,
                       a, b, thr, h0, out_h, out_s, T, D);
}